// WeaveGather_76063870812665
// MI455X (gfx1250) — compile-verified
//
#include <hip/hip_runtime.h>

typedef __attribute__((ext_vector_type(2))) float v2f;
typedef __attribute__((ext_vector_type(8))) float v8f;

#define NFEAT  128
#define NGAUSS 11
#define KDIM   (NFEAT * NGAUSS)   // 1408

// ---------------------------------------------------------------------------
// Stage 1+2 fused: gaussian membership histogram + segment sum.
// atom_split is sorted, so segment s owns a contiguous atom range found by
// binary search. One block (128 threads) per segment; thread t owns feature t
// and keeps its 11 bucket accumulators in registers -> deterministic, zero
// atomics, and the 2.25 GB expanded tensor is never materialized.
// ---------------------------------------------------------------------------
__device__ __forceinline__ int lower_bound_i32(const int* __restrict__ arr,
                                               int n, int val) {
    int lo = 0, hi = n;
    while (lo < hi) {
        int mid = (lo + hi) >> 1;
        if (arr[mid] < val) lo = mid + 1; else hi = mid;
    }
    return lo;
}

__global__ void hist_segsum_kernel(const float* __restrict__ feats,  // [N, 128]
                                   const int*   __restrict__ split,  // [N] sorted
                                   float*       __restrict__ pooled, // [B, 1408]
                                   int n_atoms)
{
    const float MU[NGAUSS] = {-1.645f, -1.08f, -0.739f, -0.468f, -0.228f, 0.0f,
                               0.228f,  0.468f, 0.739f,  1.08f,   1.645f};
    const float IS[NGAUSS] = {3.533569f, 5.882353f, 7.462687f, 8.474576f, 8.771930f,
                              8.771930f, 8.771930f, 8.474576f, 7.462687f, 5.882353f,
                              3.533569f};  // 1/sigma

    const int s = blockIdx.x;     // segment id
    const int f = threadIdx.x;    // feature id (0..127)

    const int lo = lower_bound_i32(split, n_atoms, s);
    const int hi = lower_bound_i32(split, n_atoms, s + 1);

    float acc[NGAUSS];
#pragma unroll
    for (int k = 0; k < NGAUSS; ++k) acc[k] = 0.0f;

    for (int a = lo; a < hi; ++a) {
        const float x = feats[(size_t)a * NFEAT + f];
        float e[NGAUSS];
        float sum = 0.0f;
#pragma unroll
        for (int k = 0; k < NGAUSS; ++k) {
            const float z = (x - MU[k]) * IS[k];
            e[k] = __expf(-0.5f * z * z);           // v_exp_f32 (TRANS)
            sum += e[k];
        }
        const float inv = __builtin_amdgcn_rcpf(sum); // v_rcp_f32, skip div seq
#pragma unroll
        for (int k = 0; k < NGAUSS; ++k) acc[k] += e[k] * inv;
    }

    float* dst = pooled + (size_t)s * KDIM + (size_t)f * NGAUSS;
#pragma unroll
    for (int k = 0; k < NGAUSS; ++k) dst[k] = acc[k];
}

// ---------------------------------------------------------------------------
// Pack W [K,128] into WMMA B-fragment order so each lane's fragment is one
// contiguous b64 load:  Wpack[(kk*8 + tile)*32 + lane] = {W[4kk+koff][n],
// W[4kk+koff+1][n]} with n = tile*16 + (lane&15), koff = (lane>>4)*2.
// 721 KB total -> stays L2-resident for the whole GEMM.
// ---------------------------------------------------------------------------
__global__ void pack_W_kernel(const float* __restrict__ W,   // [K, 128]
                              v2f*         __restrict__ Wpack,
                              int K)
{
    const int idx   = blockIdx.x * blockDim.x + threadIdx.x;
    const int total = (K >> 2) << 8;                  // (K/4)*8*32
    if (idx >= total) return;
    const int lane = idx & 31;
    const int tile = (idx >> 5) & 7;
    const int kk   = idx >> 8;
    const int mr   = lane & 15;
    const int koff = (lane >> 4) << 1;
    const int row  = (kk << 2) + koff;
    const int col  = (tile << 4) + mr;
    v2f v;
    v.x = W[(size_t)row * NFEAT + col];
    v.y = W[(size_t)(row + 1) * NFEAT + col];
    Wpack[idx] = v;
}

// ---------------------------------------------------------------------------
// Stage 3: out = tanh(pooled @ W + b) with V_WMMA_F32_16X16X4_F32 (full f32).
// Each wave owns a 16(M) x 64(N) strip (4 C tiles) -> 2048 waves. Depth-4
// guard-free software pipeline: steady-state loop refills unconditionally
// (last PF stages drained in a load-free epilogue), with two running cursors
// and all intra-iteration addressing folded into immediate offsets so the
// body is 16 WMMAs + 20 clause-batched b64 loads + 2 pointer adds.
// ---------------------------------------------------------------------------
__global__ void gemm_tanh_wmma_kernel(const float* __restrict__ pooled, // [B, K]
                                      const v2f*   __restrict__ Wpack,
                                      const float* __restrict__ bias,   // [128]
                                      float*       __restrict__ out,    // [B, 128]
                                      int K)
{
    constexpr int NT = 4;   // C tiles per wave (16x64 strip)
    constexpr int PF = 4;   // software pipeline depth (k-steps in flight)

    const int lane  = threadIdx.x & 31;
    const int wave  = threadIdx.x >> 5;      // 8 waves/block
    const int mtile = wave >> 1;             // 0..3
    const int nhalf = wave & 1;              // 0..1
    const int m0    = (blockIdx.x * 4 + mtile) * 16;

    const int mr   = lane & 15;              // M row (A) / N col (B) in tile
    const int koff = (lane >> 4) << 1;       // 0 or 2
    const int NK   = K >> 2;                 // 352 k-steps (K % 16 == 0 assumed)

    v8f acc[NT] = {};

    // load cursors (advance once per outer iteration)
    const float* al = pooled + (size_t)(m0 + mr) * K + koff;
    const v2f*   wl = Wpack + lane + (size_t)nhalf * NT * 32;

    v2f a_buf[PF];
    v2f b_buf[PF][NT];

    // prologue: fill the pipeline with k-steps 0..PF-1
#pragma unroll
    for (int p = 0; p < PF; ++p) {
        a_buf[p] = *(const v2f*)(al + p * 4);
#pragma unroll
        for (int t = 0; t < NT; ++t)
            b_buf[p][t] = wl[p * 256 + t * 32];
    }
    al += PF * 4;      // -> k-step PF
    wl += PF * 256;    // -> k-step PF

    // steady state: compute kk+p, refill with kk+p+PF (always in range)
    for (int kk = 0; kk + 2 * PF <= NK; kk += PF) {
#pragma unroll
        for (int p = 0; p < PF; ++p) {
#pragma unroll
            for (int t = 0; t < NT; ++t)
                acc[t] = __builtin_amdgcn_wmma_f32_16x16x4_f32(
                    /*neg_a=*/false, a_buf[p], /*neg_b=*/false, b_buf[p][t],
                    /*c_mod=*/(short)0, acc[t],
                    /*reuse_a=*/false, /*reuse_b=*/false);
            a_buf[p] = *(const v2f*)(al + p * 4);        // imm offset 16B*p
#pragma unroll
            for (int t = 0; t < NT; ++t)
                b_buf[p][t] = wl[p * 256 + t * 32];      // imm offset p*2048+t*256
        }
        al += PF * 4;
        wl += PF * 256;
    }

    // drain the last PF stages (no more loads)
#pragma unroll
    for (int p = 0; p < PF; ++p)
#pragma unroll
        for (int t = 0; t < NT; ++t)
            acc[t] = __builtin_amdgcn_wmma_f32_16x16x4_f32(
                false, a_buf[p], false, b_buf[p][t],
                (short)0, acc[t], false, false);

    // epilogue: C VGPR j -> row m0+j (lanes 0-15), row m0+j+8 (lanes 16-31)
    const int rowoff = (lane >> 4) * 8;
#pragma unroll
    for (int t = 0; t < NT; ++t) {
        const int n  = (nhalf * NT + t) * 16 + mr;
        const float bb = bias[n];
#pragma unroll
        for (int j = 0; j < 8; ++j) {
            const int row = m0 + j + rowoff;
            out[(size_t)row * NFEAT + n] = tanhf(acc[t][j] + bb);
        }
    }
}

// ---------------------------------------------------------------------------
extern "C" void kernel_launch(void* const* d_in, const int* in_sizes, int n_in,
                              void* d_out, int out_size, void* d_ws, size_t ws_size,
                              hipStream_t stream) {
    (void)n_in; (void)ws_size;

    const float* feats = (const float*)d_in[0];   // [N_ATOMS, 128] f32
    const int*   split = (const int*)  d_in[1];   // [N_ATOMS] i32, sorted
    const float* W     = (const float*)d_in[2];   // [1408, 128] f32
    const float* bias  = (const float*)d_in[3];   // [128] f32
    float*       out   = (float*)d_out;           // [B, 128] f32

    const int n_atoms = in_sizes[1];
    const int B       = out_size / NFEAT;         // 16384 segments
    const int K       = in_sizes[2] / NFEAT;      // 1408

    float* pooled = (float*)d_ws;                           // 92.3 MB
    v2f*   Wpack  = (v2f*)(pooled + (size_t)B * K);         // +721 KB

    // Pack W into fragment order (independent of hist kernel, same stream).
    const int pack_total = (K >> 2) << 8;
    hipLaunchKernelGGL(pack_W_kernel, dim3((pack_total + 255) / 256), dim3(256),
                       0, stream, W, Wpack, K);

    // Stage 1+2: one block per segment, 128 threads (4 waves).
    hipLaunchKernelGGL(hist_segsum_kernel, dim3(B), dim3(NFEAT), 0, stream,
                       feats, split, pooled, n_atoms);

    // Stage 3: 256 blocks x 256 threads = 2048 waves, 4 C tiles each.
    hipLaunchKernelGGL(gemm_tanh_wmma_kernel, dim3(B / 64), dim3(256), 0, stream,
                       pooled, Wpack, bias, out, K);
}